// EquivariantPosUpdate_28913719837200
// MI455X (gfx1250) — compile-verified
//
#include <hip/hip_runtime.h>

// ---------------------------------------------------------------------------
// EquivariantPosUpdate fused kernel for gfx1250 (MI455X, wave32, WMMA fp32)
//
// N=20000 nodes, E=320000 edges, G=32 graphs
// node feat = 32 scalars + 16 vec (80 f32), edge feat = 16 + 8 vec (40 f32)
//
// Dead code: only normed[:, :S] is consumed downstream -> the vector outputs
// of the SECOND fctp (We011/We101/We111) and ada_norm's vector RMS branch are
// skipped entirely.
//
// All dense channel-mixing contractions run on V_WMMA_F32_16X16X4_F32 (full
// fp32). Each wave owns two 16-edge M-tiles and processes BOTH inside each
// K-step so every B (weight) fragment is loaded once and feeds two WMMAs
// (B-reuse across M; halves weight-load instruction traffic in the K=256
// loops). Remaining VALU per edge: small eps-contractions, vector channel
// mixes, adaLN, force + segment_sum atomics.
// ---------------------------------------------------------------------------

#define NN 20000
#define EE 320000
#define GG 32
#define DN 80   // node feature floats
#define DE 40   // edge feature floats

typedef float v2f __attribute__((ext_vector_type(2)));
typedef float v8f __attribute__((ext_vector_type(8)));

// rescale constants
#define RS32   0.17677669529663689f   // 1/sqrt(32)
#define RS16   0.25f                  // 1/sqrt(16)
#define RS8    0.35355339059327373f   // 1/sqrt(8)
#define S320   0.05590169943749474f   // 1/sqrt(320)  (both fctp fan-ins)
#define C3S320 0.03227486121839514f   // (1/sqrt(3))/sqrt(320)
#define C6S320 0.02282177322938192f   // (1/sqrt(6))/sqrt(320)

#define WMMA_F32X4(c, a, b) \
  (c) = __builtin_amdgcn_wmma_f32_16x16x4_f32(false, (a), false, (b), (short)0, (c), false, false)

#define V8F_ZERO {0.f, 0.f, 0.f, 0.f, 0.f, 0.f, 0.f, 0.f}

// GEMM tile pair: out{0,1}[16 edges][16] = A{0,1}(16xK) * W^T(Kx16)*scale + bias
// B fragments loaded once per K-step, shared by both M-tiles.
__device__ __forceinline__ void lin_tile_pair(const float* __restrict__ a0r,
                                              const float* __restrict__ a1r, int K,
                                              const float* __restrict__ W, float scale,
                                              const float* __restrict__ bias,
                                              float* __restrict__ out0,
                                              float* __restrict__ out1, int lane) {
  const int mi = lane & 15;
  const int kh = (lane >> 4) << 1;
  v8f c0 = V8F_ZERO, c1 = V8F_ZERO;
  for (int kb = 0; kb < K; kb += 4) {
    const int k0 = kb + kh;
    v2f b;  b.x  = W[mi * K + k0] * scale; b.y  = W[mi * K + k0 + 1] * scale;
    v2f a0; a0.x = a0r[k0];                a0.y = a0r[k0 + 1];
    v2f a1; a1.x = a1r[k0];                a1.y = a1r[k0 + 1];
    WMMA_F32X4(c0, a0, b);
    WMMA_F32X4(c1, a1, b);
  }
  const float bb = bias[mi];
  const int hr = (lane >> 4) << 3;
#pragma unroll
  for (int r = 0; r < 8; ++r) {
    out0[(hr + r) * 16 + mi] = c0[r] + bb;
    out1[(hr + r) * 16 + mi] = c1[r] + bb;
  }
}

// Channel-mixing GEMM tile pair: out{0,1}[16 edges][outW] = A{0,1}(16xK)*B(KxoutW)
// A element stride astr (1 = contiguous scalars, 3 = one vector component).
// bval(k, j) returns the (already scaled) B element; loaded once per K-step.
template <typename BF>
__device__ __forceinline__ void mix_tile_pair(const float* __restrict__ a0r,
                                              const float* __restrict__ a1r, int astr,
                                              int K, BF bval,
                                              float* __restrict__ out0,
                                              float* __restrict__ out1, int outW,
                                              int lane) {
  const int mi = lane & 15;
  const int kh = (lane >> 4) << 1;
  const int hr = (lane >> 4) << 3;
  const int NT = outW >> 4;
  for (int nt = 0; nt < NT; ++nt) {
    const int j = nt * 16 + mi;
    v8f c0 = V8F_ZERO, c1 = V8F_ZERO;
    for (int kb = 0; kb < K; kb += 4) {
      const int k0 = kb + kh;
      v2f b;  b.x  = bval(k0, j);       b.y  = bval(k0 + 1, j);
      v2f a0; a0.x = a0r[k0 * astr];    a0.y = a0r[(k0 + 1) * astr];
      v2f a1; a1.x = a1r[k0 * astr];    a1.y = a1r[(k0 + 1) * astr];
      WMMA_F32X4(c0, a0, b);
      WMMA_F32X4(c1, a1, b);
    }
#pragma unroll
    for (int r = 0; r < 8; ++r) {
      out0[(hr + r) * outW + nt * 16 + mi] = c0[r];
      out1[(hr + r) * outW + nt * 16 + mi] = c1[r];
    }
  }
}

// fctp scalar output, both tiles of the wave's 32-edge region:
//   o0 = ( outer(x0,y0) @ W000 + c3 * vdot(x1,y1) @ W110 ) / sqrt(320) + bias
// Region base pointers: x0b/y0b = &buf[te0][0] (stride 16), xvb/yvb (stride 24).
__device__ __forceinline__ void fctp0_tile_pair(const float* __restrict__ x0b,
                                                const float* __restrict__ y0b,
                                                const float* __restrict__ xvb,
                                                const float* __restrict__ yvb,
                                                const float* __restrict__ W000,
                                                const float* __restrict__ W110,
                                                const float* __restrict__ bias,
                                                float* __restrict__ outb, int lane) {
  const int mi = lane & 15;
  const int kh = (lane >> 4) << 1;
  const float* x0A = x0b + mi * 16;        const float* x0B = x0b + (16 + mi) * 16;
  const float* y0A = y0b + mi * 16;        const float* y0B = y0b + (16 + mi) * 16;
  const float* xvA = xvb + mi * 24;        const float* xvB = xvb + (16 + mi) * 24;
  const float* yvA = yvb + mi * 24;        const float* yvB = yvb + (16 + mi) * 24;
  v8f c0 = V8F_ZERO, c1 = V8F_ZERO;
  // outer-product term, K = 256 (k = u*16+v); k0 even -> (k0&15)+1 <= 15
  for (int kb = 0; kb < 256; kb += 4) {
    const int k0 = kb + kh;
    const int v0 = k0 & 15;
    v2f b;  b.x = W000[k0 * 16 + mi] * S320;  b.y = W000[(k0 + 1) * 16 + mi] * S320;
    const float xuA = x0A[k0 >> 4], xuB = x0B[k0 >> 4];
    v2f a0; a0.x = xuA * y0A[v0];             a0.y = xuA * y0A[v0 + 1];
    v2f a1; a1.x = xuB * y0B[v0];             a1.y = xuB * y0B[v0 + 1];
    WMMA_F32X4(c0, a0, b);
    WMMA_F32X4(c1, a1, b);
  }
  // vector-dot term, K = 64 (k = u*8+v)
  for (int kb = 0; kb < 64; kb += 4) {
    const int k0 = kb + kh;
    const int uo = (k0 >> 3) * 3, vo = (k0 & 7) * 3;
    v2f b;  b.x = W110[k0 * 16 + mi] * C3S320; b.y = W110[(k0 + 1) * 16 + mi] * C3S320;
    const float* xa = xvA + uo; const float* yb = yvA + vo;
    v2f a0;
    a0.x = xa[0] * yb[0] + xa[1] * yb[1] + xa[2] * yb[2];
    a0.y = xa[0] * yb[3] + xa[1] * yb[4] + xa[2] * yb[5];
    const float* xc = xvB + uo; const float* yd = yvB + vo;
    v2f a1;
    a1.x = xc[0] * yd[0] + xc[1] * yd[1] + xc[2] * yd[2];
    a1.y = xc[0] * yd[3] + xc[1] * yd[4] + xc[2] * yd[5];
    WMMA_F32X4(c0, a0, b);
    WMMA_F32X4(c1, a1, b);
  }
  const float bb = bias[mi];
  const int hr = (lane >> 4) << 3;
#pragma unroll
  for (int r = 0; r < 8; ++r) {
    outb[(hr + r) * 16 + mi]        = c0[r] + bb;
    outb[(16 + hr + r) * 16 + mi]   = c1[r] + bb;
  }
}

// ---------------------------------------------------------------------------
__global__ void zero_kernel(float* __restrict__ p, int n) {
  int i = blockIdx.x * blockDim.x + threadIdx.x;
  if (i < n) p[i] = 0.f;
}

// ada = silu(t) @ W_ada + b_ada ; only first 32 of 40 columns are live.
__global__ void ada_kernel(const float* __restrict__ t, const float* __restrict__ W_ada,
                           const float* __restrict__ b_ada, float* __restrict__ out) {
  const int g = blockIdx.x;       // G blocks
  const int j = threadIdx.x;      // 32 threads
  float acc = b_ada[j];
  for (int k = 0; k < 128; ++k) {
    const float tv = t[g * 128 + k];
    acc += (tv / (1.f + __expf(-tv))) * W_ada[k * 40 + j];
  }
  out[g * 32 + j] = acc;
}

// ---------------------------------------------------------------------------
__global__ __launch_bounds__(64) void fused_edge_kernel(
    const float* __restrict__ h_node, const float* __restrict__ h_edge,
    const int* __restrict__ ei, const float* __restrict__ rvec,
    const float* __restrict__ dist, const int* __restrict__ batch,
    const float* __restrict__ W_src0, const float* __restrict__ W_src1, const float* __restrict__ b_src,
    const float* __restrict__ W_dst0, const float* __restrict__ W_dst1, const float* __restrict__ b_dst,
    const float* __restrict__ Wn000, const float* __restrict__ Wn110,
    const float* __restrict__ Wn011, const float* __restrict__ Wn101,
    const float* __restrict__ Wn111, const float* __restrict__ bn0,
    const float* __restrict__ Wnt0, const float* __restrict__ Wnt1, const float* __restrict__ bnt,
    const float* __restrict__ Wet0, const float* __restrict__ Wet1, const float* __restrict__ bet,
    const float* __restrict__ We000, const float* __restrict__ We110, const float* __restrict__ be0,
    const float* __restrict__ Wp1, const float* __restrict__ bp1,
    const float* __restrict__ Wp2, const float* __restrict__ bp2,
    const float* __restrict__ adaBuf, float* __restrict__ out) {
  // LDS regions, 64 edges / block, explicit lifetime reuse:
  __shared__ float R0[64][16];  // src scalar feats      -> fused_trans scalar
  __shared__ float R1[64][16];  // dst scalar feats      -> edge_trans scalar
  __shared__ float R2[64][24];  // src vector feats      -> fused_trans vector
  __shared__ float R3[64][24];  // dst vector feats      -> edge_trans vector
  __shared__ float R4[64][16];  // fctp1 scalar out      -> fctp2 scalar / normed
  __shared__ float R6[64][64];  // M011 panel -> Q_i panel -> MLP hidden (32 cols)
  __shared__ float R7[64][64];  // M101 panel

  const int tid  = threadIdx.x;
  const int lane = tid & 31;
  const int wv   = tid >> 5;                 // wave id 0..1
  const int mi   = lane & 15;
  const int kh   = (lane >> 4) << 1;
  const int te0  = wv * 32;                  // wave's 32-edge region
  const int e0   = blockIdx.x * 64;
  const int e    = e0 + tid;                 // this thread's edge (VALU stages)
  const int eb0  = e0 + te0;
  const int srcN = ei[e];
  const int dstN = ei[EE + e];

  // ===== stage 1: input linears (WMMA pairs) + vector mixes (VALU) =========
  {
    const int nsA = ei[eb0 + mi],       nsB = ei[eb0 + 16 + mi];
    const int ndA = ei[EE + eb0 + mi],  ndB = ei[EE + eb0 + 16 + mi];
    lin_tile_pair(h_node + nsA * DN, h_node + nsB * DN, 32, W_src0, RS32, b_src,
                  &R0[te0][0], &R0[te0 + 16][0], lane);
    lin_tile_pair(h_node + ndA * DN, h_node + ndB * DN, 32, W_dst0, RS32, b_dst,
                  &R1[te0][0], &R1[te0 + 16][0], lane);
  }
  {
    float aS[24], aD[24];
#pragma unroll
    for (int j = 0; j < 24; ++j) { aS[j] = 0.f; aD[j] = 0.f; }
    for (int v = 0; v < 16; ++v) {
      const float sx = h_node[srcN * DN + 32 + v * 3 + 0];
      const float sy = h_node[srcN * DN + 32 + v * 3 + 1];
      const float sz = h_node[srcN * DN + 32 + v * 3 + 2];
      const float dx = h_node[dstN * DN + 32 + v * 3 + 0];
      const float dy = h_node[dstN * DN + 32 + v * 3 + 1];
      const float dz = h_node[dstN * DN + 32 + v * 3 + 2];
#pragma unroll
      for (int w = 0; w < 8; ++w) {
        const float ws_ = W_src1[w * 16 + v];
        const float wd_ = W_dst1[w * 16 + v];
        aS[w * 3 + 0] += ws_ * sx; aS[w * 3 + 1] += ws_ * sy; aS[w * 3 + 2] += ws_ * sz;
        aD[w * 3 + 0] += wd_ * dx; aD[w * 3 + 1] += wd_ * dy; aD[w * 3 + 2] += wd_ * dz;
      }
    }
#pragma unroll
    for (int j = 0; j < 24; ++j) { R2[tid][j] = aS[j] * RS16; R3[tid][j] = aD[j] * RS16; }
  }
  __syncthreads();

  // ===== stage 2a: fctp1 scalar + W011/W101 mix panels (WMMA pairs) ========
  fctp0_tile_pair(&R0[te0][0], &R1[te0][0], &R2[te0][0], &R3[te0][0],
                  Wn000, Wn110, bn0, &R4[te0][0], lane);
  // M011[e, v*8+w] = c3/sqrt(320) * sum_u x0[e,u] * Wn011[u,v,w]
  mix_tile_pair(&R0[te0 + mi][0], &R0[te0 + 16 + mi][0], 1, 16,
                [&](int k, int j) { return Wn011[((k << 3) + (j >> 3)) * 8 + (j & 7)] * C3S320; },
                &R6[te0][0], &R6[te0 + 16][0], 64, lane);
  // M101[e, u*8+w] = c3/sqrt(320) * sum_v y0[e,v] * Wn101[u,v,w]
  mix_tile_pair(&R1[te0 + mi][0], &R1[te0 + 16 + mi][0], 1, 16,
                [&](int k, int j) { return Wn101[((j >> 3) * 16 + k) * 8 + (j & 7)] * C3S320; },
                &R7[te0][0], &R7[te0 + 16][0], 64, lane);
  __syncthreads();

  // ===== stage 2b: fctp1 vector output, part 1 (M011/M101 contraction) =====
  float o1[24];                               // fctp1 vector out, stays in regs
  {
#pragma unroll
    for (int j = 0; j < 24; ++j) o1[j] = 0.f;
    const float* m011 = &R6[tid][0];
    const float* m101 = &R7[tid][0];
    const float* xv = &R2[tid][0];
    const float* yv = &R3[tid][0];
    for (int v = 0; v < 8; ++v) {
      const float yx = yv[v * 3 + 0], yy = yv[v * 3 + 1], yz = yv[v * 3 + 2];
#pragma unroll
      for (int w = 0; w < 8; ++w) {
        const float sc = m011[v * 8 + w];
        o1[w * 3 + 0] += sc * yx; o1[w * 3 + 1] += sc * yy; o1[w * 3 + 2] += sc * yz;
      }
    }
    for (int u = 0; u < 8; ++u) {
      const float xx = xv[u * 3 + 0], xy = xv[u * 3 + 1], xz = xv[u * 3 + 2];
#pragma unroll
      for (int w = 0; w < 8; ++w) {
        const float sc = m101[u * 8 + w];
        o1[w * 3 + 0] += sc * xx; o1[w * 3 + 1] += sc * xy; o1[w * 3 + 2] += sc * xz;
      }
    }
  }
  __syncthreads();   // WAR: Q panels reuse R6

  // ===== stage 2c: W111 term via per-component Q panels =====================
  // Q_i[e, v*8+w] = c6/sqrt(320) * sum_u x1[e,u,i] * Wn111[u,v,w]
  // eps antisymmetry: o1[w, (i+2)%3] += Q_i*y1[v,(i+1)%3]
  //                   o1[w, (i+1)%3] -= Q_i*y1[v,(i+2)%3]
  for (int i = 0; i < 3; ++i) {
    mix_tile_pair(&R2[te0 + mi][i], &R2[te0 + 16 + mi][i], 3, 8,
                  [&](int k, int j) { return Wn111[((k << 3) + (j >> 3)) * 8 + (j & 7)] * C6S320; },
                  &R6[te0][0], &R6[te0 + 16][0], 64, lane);
    __syncthreads();
    {
      const int jp = (i + 1) % 3, kp = (i + 2) % 3;
      const float* q = &R6[tid][0];
      const float* yv = &R3[tid][0];
      for (int v = 0; v < 8; ++v) {
        const float yjp = yv[v * 3 + jp];
        const float ykp = yv[v * 3 + kp];
#pragma unroll
        for (int w = 0; w < 8; ++w) {
          const float qq = q[v * 8 + w];
          o1[w * 3 + kp] += qq * yjp;
          o1[w * 3 + jp] -= qq * ykp;
        }
      }
    }
    __syncthreads();   // WAR before next Q panel / before stage 3 writes
  }

  // ===== stage 3: mid linears (WMMA pairs scalar, VALU vector) ==============
  lin_tile_pair(&R4[te0 + mi][0], &R4[te0 + 16 + mi][0], 16, Wnt0, RS16, bnt,
                &R0[te0][0], &R0[te0 + 16][0], lane);
  lin_tile_pair(h_edge + (eb0 + mi) * DE, h_edge + (eb0 + 16 + mi) * DE, 16, Wet0, RS16, bet,
                &R1[te0][0], &R1[te0 + 16][0], lane);
  {
    float oT[24], oE[24];
#pragma unroll
    for (int j = 0; j < 24; ++j) { oT[j] = 0.f; oE[j] = 0.f; }
    const float* hev = h_edge + e * DE + 16;
    for (int v = 0; v < 8; ++v) {
      const float fx = o1[v * 3 + 0], fy = o1[v * 3 + 1], fz = o1[v * 3 + 2];
      const float gx = hev[v * 3 + 0], gy = hev[v * 3 + 1], gz = hev[v * 3 + 2];
#pragma unroll
      for (int w = 0; w < 8; ++w) {
        const float wt = Wnt1[w * 8 + v];
        const float we = Wet1[w * 8 + v];
        oT[w * 3 + 0] += wt * fx; oT[w * 3 + 1] += wt * fy; oT[w * 3 + 2] += wt * fz;
        oE[w * 3 + 0] += we * gx; oE[w * 3 + 1] += we * gy; oE[w * 3 + 2] += we * gz;
      }
    }
#pragma unroll
    for (int j = 0; j < 24; ++j) { R2[tid][j] = oT[j] * RS8; R3[tid][j] = oE[j] * RS8; }
  }
  __syncthreads();

  // ===== stage 4: fctp2 — only scalar output is live downstream ============
  fctp0_tile_pair(&R0[te0][0], &R1[te0][0], &R2[te0][0], &R3[te0][0],
                  We000, We110, be0, &R4[te0][0], lane);
  __syncthreads();

  // ===== stage 5: adaLN on scalars (vector branch is dead code) ============
  {
    float x[16];
    float mu = 0.f;
#pragma unroll
    for (int w = 0; w < 16; ++w) { x[w] = R4[tid][w]; mu += x[w]; }
    mu *= (1.f / 16.f);
    float var = 0.f;
#pragma unroll
    for (int w = 0; w < 16; ++w) { const float d = x[w] - mu; var += d * d; }
    var *= (1.f / 16.f);
    const float inv = rsqrtf(var + 1e-5f);
    const int g = batch[srcN];
    const float* ad = adaBuf + g * 32;
#pragma unroll
    for (int w = 0; w < 16; ++w)
      R4[tid][w] = (x[w] - mu) * inv * (1.f + ad[w]) + ad[16 + w];
  }
  __syncthreads();

  // ===== stage 6: MLP layer 1 (K=16, N=32, silu) via WMMA pairs ============
  {
    const float* a0r = &R4[te0 + mi][0];
    const float* a1r = &R4[te0 + 16 + mi][0];
#pragma unroll
    for (int h2 = 0; h2 < 2; ++h2) {
      v8f c0 = V8F_ZERO, c1 = V8F_ZERO;
      for (int kb = 0; kb < 16; kb += 4) {
        const int k0 = kb + kh;
        v2f b;  b.x  = Wp1[(h2 * 16 + mi) * 16 + k0] * RS16;
                b.y  = Wp1[(h2 * 16 + mi) * 16 + k0 + 1] * RS16;
        v2f a0; a0.x = a0r[k0]; a0.y = a0r[k0 + 1];
        v2f a1; a1.x = a1r[k0]; a1.y = a1r[k0 + 1];
        WMMA_F32X4(c0, a0, b);
        WMMA_F32X4(c1, a1, b);
      }
      const float bb = bp1[h2 * 16 + mi];
      const int hr = (lane >> 4) << 3;
#pragma unroll
      for (int r = 0; r < 8; ++r) {
        float v0 = c0[r] + bb;  v0 = v0 / (1.f + __expf(-v0));
        float v1 = c1[r] + bb;  v1 = v1 / (1.f + __expf(-v1));
        R6[te0 + hr + r][h2 * 16 + mi]      = v0;   // MLP hidden in first 32 cols
        R6[te0 + 16 + hr + r][h2 * 16 + mi] = v1;
      }
    }
  }
  __syncthreads();

  // ===== stage 7: scalar weight, force, segment_sum (atomics) ==============
  {
    float sw = 0.f;
#pragma unroll
    for (int j = 0; j < 32; ++j) sw += R6[tid][j] * Wp2[j];
    sw = sw * RS32 + bp2[0];                // 1/sqrt(32)
    const float dd = dist[e];
    const float f = sw / (dd * (dd + 1.f));
    atomicAdd(&out[srcN * 3 + 0], f * rvec[e * 3 + 0]);
    atomicAdd(&out[srcN * 3 + 1], f * rvec[e * 3 + 1]);
    atomicAdd(&out[srcN * 3 + 2], f * rvec[e * 3 + 2]);
  }
}

// ---------------------------------------------------------------------------
extern "C" void kernel_launch(void* const* d_in, const int* in_sizes, int n_in,
                              void* d_out, int out_size, void* d_ws, size_t ws_size,
                              hipStream_t stream) {
  const float* h_node = (const float*)d_in[0];
  const float* h_edge = (const float*)d_in[1];
  // d_in[2] = pos (unused by the reference output)
  const int*   ei     = (const int*)d_in[3];
  const float* rvec   = (const float*)d_in[4];
  const float* dist   = (const float*)d_in[5];
  const float* t      = (const float*)d_in[6];
  const int*   batch  = (const int*)d_in[7];
  const float* W_src0 = (const float*)d_in[8];
  const float* W_src1 = (const float*)d_in[9];
  const float* b_src  = (const float*)d_in[10];
  const float* W_dst0 = (const float*)d_in[11];
  const float* W_dst1 = (const float*)d_in[12];
  const float* b_dst  = (const float*)d_in[13];
  const float* Wn000  = (const float*)d_in[14];
  const float* Wn110  = (const float*)d_in[15];
  const float* Wn011  = (const float*)d_in[16];
  const float* Wn101  = (const float*)d_in[17];
  const float* Wn111  = (const float*)d_in[18];
  const float* bn0    = (const float*)d_in[19];
  const float* Wnt0   = (const float*)d_in[20];
  const float* Wnt1   = (const float*)d_in[21];
  const float* bnt    = (const float*)d_in[22];
  const float* Wet0   = (const float*)d_in[23];
  const float* Wet1   = (const float*)d_in[24];
  const float* bet    = (const float*)d_in[25];
  const float* We000  = (const float*)d_in[26];
  const float* We110  = (const float*)d_in[27];
  // d_in[28..30] = We011/We101/We111 : provably dead (vector branch unused)
  const float* be0    = (const float*)d_in[31];
  const float* W_ada  = (const float*)d_in[32];
  const float* b_ada  = (const float*)d_in[33];
  const float* Wp1    = (const float*)d_in[34];
  const float* bp1    = (const float*)d_in[35];
  const float* Wp2    = (const float*)d_in[36];
  const float* bp2    = (const float*)d_in[37];

  float* adaBuf = (float*)d_ws;            // G*32 floats
  float* out    = (float*)d_out;           // N*3 floats

  zero_kernel<<<(NN * 3 + 255) / 256, 256, 0, stream>>>(out, NN * 3);
  ada_kernel<<<GG, 32, 0, stream>>>(t, W_ada, b_ada, adaBuf);
  fused_edge_kernel<<<EE / 64, 64, 0, stream>>>(
      h_node, h_edge, ei, rvec, dist, batch,
      W_src0, W_src1, b_src, W_dst0, W_dst1, b_dst,
      Wn000, Wn110, Wn011, Wn101, Wn111, bn0,
      Wnt0, Wnt1, bnt, Wet0, Wet1, bet,
      We000, We110, be0, Wp1, bp1, Wp2, bp2,
      adaBuf, out);
}